// GCN_69260642615660
// MI455X (gfx1250) — compile-verified
//
#include <hip/hip_runtime.h>

typedef __attribute__((ext_vector_type(2))) float v2f;
typedef __attribute__((ext_vector_type(8))) float v8f;

#define NFEAT 512
#define NHID  128
#define NCLS  40
#define NCLSP 48   // NCLS padded to multiple of 16 for WMMA

// ---------------- init: deg=1 (self loop), agg1=0, out=b2 ----------------
__global__ __launch_bounds__(256) void gcn_init_kernel(float* __restrict__ deg,
                                                       float* __restrict__ agg1,
                                                       float* __restrict__ out,
                                                       const float* __restrict__ b2,
                                                       int n) {
  long long idx = (long long)blockIdx.x * blockDim.x + threadIdx.x;
  if (idx < n) deg[idx] = 1.0f;                         // self-loop weight
  if (idx < (long long)n * NCLS) out[idx] = b2[idx % NCLS];
  if (idx < (long long)n * NHID) agg1[idx] = 0.0f;
}

// ---------------- degree over targets ----------------
__global__ __launch_bounds__(256) void gcn_deg_kernel(const int* __restrict__ col,
                                                      const float* __restrict__ w,
                                                      float* __restrict__ deg, int E) {
  int e = blockIdx.x * blockDim.x + threadIdx.x;
  if (e < E) atomicAdd(&deg[col[e]], w[e]);
}

__global__ __launch_bounds__(256) void gcn_dis_kernel(float* __restrict__ deg, int n) {
  int i = blockIdx.x * blockDim.x + threadIdx.x;
  if (i < n) {
    float d = deg[i];
    deg[i] = d > 0.0f ? rsqrtf(d) : 0.0f;
  }
}

// ---------------- GEMM1: h = x @ W1   [n,512] x [512,128] ----------------
// Block = 256 threads (8 waves). Block computes 16 rows x 128 cols.
// Wave w owns col tile [16w, 16w+16). K stepped by 4 via v_wmma_f32_16x16x4_f32.
__global__ __launch_bounds__(256) void gcn_gemm1_kernel(const float* __restrict__ x,
                                                        const float* __restrict__ W1,
                                                        float* __restrict__ h, int n) {
  __shared__ float As[16][NFEAT + 4];   // +4 pad -> bank = (4m+k)%64, conflict-free
  const int rowBase = blockIdx.x * 16;
  const int tid = threadIdx.x;
  const bool fullTile = (rowBase + 16 <= n);

  if (fullTile) {
    // vectorized staging: 16*128 float4, 8 per thread, fully coalesced
    const float4* x4 = (const float4*)(x + (long long)rowBase * NFEAT);
    for (int i = tid; i < 16 * (NFEAT / 4); i += 256) {
      int r = i >> 7, c4 = i & 127;
      *(float4*)&As[r][c4 * 4] = x4[(long long)r * (NFEAT / 4) + c4];
    }
  } else {
    for (int i = tid; i < 16 * NFEAT; i += 256) {
      int r = i >> 9, cc = i & (NFEAT - 1);
      int gr = rowBase + r;
      As[r][cc] = (gr < n) ? x[(long long)gr * NFEAT + cc] : 0.0f;
    }
  }
  __syncthreads();

  const int wave = tid >> 5, lane = tid & 31;
  const int colBase = wave * 16;
  const int m = lane & 15;                 // A row / B,C col within tile
  const int koff = (lane < 16) ? 0 : 2;    // per ISA 16x4 f32 A layout

  v8f acc = {};
  for (int k = 0; k < NFEAT; k += 4) {
    v2f a, b;
    a.x = As[m][k + koff];
    a.y = As[m][k + koff + 1];
    const float* Bp = W1 + (long long)(k + koff) * NHID + colBase + m;
    b.x = Bp[0];
    b.y = Bp[NHID];
    acc = __builtin_amdgcn_wmma_f32_16x16x4_f32(false, a, false, b,
                                                (short)0, acc, false, false);
  }

  const int mBase = (lane < 16) ? 0 : 8;   // C/D layout: vgpr v -> row mBase+v
  if (fullTile) {
    float* hp = h + (long long)(rowBase + mBase) * NHID + colBase + m;
#pragma unroll
    for (int v = 0; v < 8; ++v) hp[(long long)v * NHID] = acc[v];
  } else {
#pragma unroll
    for (int v = 0; v < 8; ++v) {
      int gr = rowBase + mBase + v;
      if (gr < n) h[(long long)gr * NHID + colBase + m] = acc[v];
    }
  }
}

// ---------------- scatter layer 1: one wave per edge, 128 cols ----------------
__global__ __launch_bounds__(256) void gcn_scatter1_kernel(const int* __restrict__ rowi,
                                                           const int* __restrict__ coli,
                                                           const float* __restrict__ w,
                                                           const float* __restrict__ dis,
                                                           const float4* __restrict__ h4,
                                                           float* __restrict__ agg, int E) {
  long long t = (long long)blockIdx.x * blockDim.x + threadIdx.x;
  int e = (int)(t >> 5);
  int lane = (int)(t & 31);
  if (e >= E) return;
  int r = rowi[e], c = coli[e];
  float norm = dis[r] * w[e] * dis[c];
  float4 v = h4[(long long)r * (NHID / 4) + lane];   // coalesced 512B per wave
  float* dst = agg + (long long)c * NHID + lane * 4;
  atomicAdd(dst + 0, norm * v.x);
  atomicAdd(dst + 1, norm * v.y);
  atomicAdd(dst + 2, norm * v.z);
  atomicAdd(dst + 3, norm * v.w);
}

// ---------------- self-loop + bias + relu (in place on agg1) ----------------
__global__ __launch_bounds__(256) void gcn_relu_kernel(float* __restrict__ agg,
                                                       const float* __restrict__ hxw1,
                                                       const float* __restrict__ dis,
                                                       const float* __restrict__ b1, int n) {
  long long i = (long long)blockIdx.x * blockDim.x + threadIdx.x;
  if (i >= (long long)n * NHID) return;
  int node = (int)(i >> 7);
  int cc = (int)(i & (NHID - 1));
  float d = dis[node];
  float v = agg[i] + d * d * hxw1[i] + b1[cc];
  agg[i] = v > 0.0f ? v : 0.0f;
}

// ---------------- GEMM2: o = h @ W2   [n,128] x [128,40->48] ----------------
// Block = 96 threads (3 waves); wave w owns col tile [16w, 16w+16).
__global__ __launch_bounds__(96) void gcn_gemm2_kernel(const float* __restrict__ h,
                                                       const float* __restrict__ W2,
                                                       float* __restrict__ o, int n) {
  __shared__ float Hs[16][NHID + 4];
  __shared__ float Bs[NHID][NCLSP];
  const int rowBase = blockIdx.x * 16;
  const int tid = threadIdx.x;
  const bool fullTile = (rowBase + 16 <= n);

  if (fullTile) {
    const float4* h4 = (const float4*)(h + (long long)rowBase * NHID);
    for (int i = tid; i < 16 * (NHID / 4); i += 96) {
      int r = i >> 5, c4 = i & 31;
      *(float4*)&Hs[r][c4 * 4] = h4[(long long)r * (NHID / 4) + c4];
    }
  } else {
    for (int i = tid; i < 16 * NHID; i += 96) {
      int r = i >> 7, cc = i & (NHID - 1);
      int gr = rowBase + r;
      Hs[r][cc] = (gr < n) ? h[(long long)gr * NHID + cc] : 0.0f;
    }
  }
  for (int i = tid; i < NHID * NCLSP; i += 96) {
    int kk = i / NCLSP, cc = i % NCLSP;
    Bs[kk][cc] = (cc < NCLS) ? W2[kk * NCLS + cc] : 0.0f;
  }
  __syncthreads();

  const int wave = tid >> 5, lane = tid & 31;
  const int colBase = wave * 16;
  const int m = lane & 15;
  const int koff = (lane < 16) ? 0 : 2;

  v8f acc = {};
  for (int k = 0; k < NHID; k += 4) {
    v2f a, b;
    a.x = Hs[m][k + koff];
    a.y = Hs[m][k + koff + 1];
    b.x = Bs[k + koff][colBase + m];
    b.y = Bs[k + koff + 1][colBase + m];
    acc = __builtin_amdgcn_wmma_f32_16x16x4_f32(false, a, false, b,
                                                (short)0, acc, false, false);
  }

  const int mBase = (lane < 16) ? 0 : 8;
  const int gc = colBase + m;
  const bool colOk = (gc < NCLS);          // uniform guard, single exec-mask set
  if (fullTile) {
    if (colOk) {
      float* op = o + (long long)(rowBase + mBase) * NCLS + gc;
#pragma unroll
      for (int v = 0; v < 8; ++v) op[(long long)v * NCLS] = acc[v];
    }
  } else {
#pragma unroll
    for (int v = 0; v < 8; ++v) {
      int gr = rowBase + mBase + v;
      if (gr < n && colOk) o[(long long)gr * NCLS + gc] = acc[v];
    }
  }
}

// ---------------- scatter layer 2: one wave per edge, 40 cols ----------------
__global__ __launch_bounds__(256) void gcn_scatter2_kernel(const int* __restrict__ rowi,
                                                           const int* __restrict__ coli,
                                                           const float* __restrict__ w,
                                                           const float* __restrict__ dis,
                                                           const float* __restrict__ h2,
                                                           float* __restrict__ out, int E) {
  long long t = (long long)blockIdx.x * blockDim.x + threadIdx.x;
  int e = (int)(t >> 5);
  int lane = (int)(t & 31);
  if (e >= E) return;
  int r = rowi[e], c = coli[e];
  float norm = dis[r] * w[e] * dis[c];
  float v0 = h2[(long long)r * NCLS + lane];
  atomicAdd(&out[(long long)c * NCLS + lane], norm * v0);
  if (lane < NCLS - 32) {
    float v1 = h2[(long long)r * NCLS + 32 + lane];
    atomicAdd(&out[(long long)c * NCLS + 32 + lane], norm * v1);
  }
}

// ---------------- layer-2 self loop (out already holds b2 + scatter) ----------------
__global__ __launch_bounds__(256) void gcn_selfloop2_kernel(float* __restrict__ out,
                                                            const float* __restrict__ h2,
                                                            const float* __restrict__ dis, int n) {
  long long i = (long long)blockIdx.x * blockDim.x + threadIdx.x;
  if (i >= (long long)n * NCLS) return;
  int node = (int)(i / NCLS);
  float d = dis[node];
  out[i] += d * d * h2[i];
}

extern "C" void kernel_launch(void* const* d_in, const int* in_sizes, int n_in,
                              void* d_out, int out_size, void* d_ws, size_t ws_size,
                              hipStream_t stream) {
  const float* x  = (const float*)d_in[0];
  const int*   ei = (const int*)d_in[1];
  const float* ew = (const float*)d_in[2];
  const float* W1 = (const float*)d_in[3];
  const float* b1 = (const float*)d_in[4];
  const float* W2 = (const float*)d_in[5];
  const float* b2 = (const float*)d_in[6];

  const int n = in_sizes[0] / NFEAT;
  const int E = in_sizes[2];
  const int* rowi = ei;        // edge_index[0] (source)
  const int* coli = ei + E;    // edge_index[1] (target)

  float* ws   = (float*)d_ws;
  float* dis  = ws;                                   // [n]  (deg -> dis in place)
  float* hxw1 = dis + n;                              // [n*128]
  float* agg1 = hxw1 + (long long)n * NHID;           // [n*128]
  float* hw2  = agg1 + (long long)n * NHID;           // [n*40]
  float* out  = (float*)d_out;                        // [n*40]

  const int B = 256;
  long long nh = (long long)n * NHID;
  long long no = (long long)n * NCLS;
  long long edgeThreads = (long long)E * 32;

  gcn_init_kernel<<<(unsigned)((nh + B - 1) / B), B, 0, stream>>>(dis, agg1, out, b2, n);
  gcn_deg_kernel<<<(E + B - 1) / B, B, 0, stream>>>(coli, ew, dis, E);
  gcn_dis_kernel<<<(n + B - 1) / B, B, 0, stream>>>(dis, n);

  gcn_gemm1_kernel<<<(n + 15) / 16, 256, 0, stream>>>(x, W1, hxw1, n);
  gcn_scatter1_kernel<<<(unsigned)((edgeThreads + B - 1) / B), B, 0, stream>>>(
      rowi, coli, ew, dis, (const float4*)hxw1, agg1, E);
  gcn_relu_kernel<<<(unsigned)((nh + B - 1) / B), B, 0, stream>>>(agg1, hxw1, dis, b1, n);

  gcn_gemm2_kernel<<<(n + 15) / 16, 96, 0, stream>>>(agg1, W2, hw2, n);
  gcn_scatter2_kernel<<<(unsigned)((edgeThreads + B - 1) / B), B, 0, stream>>>(
      rowi, coli, ew, dis, hw2, out, E);
  gcn_selfloop2_kernel<<<(unsigned)((no + B - 1) / B), B, 0, stream>>>(out, hw2, dis, n);
}